// RandomShiftsAug_6631429505767
// MI455X (gfx1250) — compile-verified
//
#include <hip/hip_runtime.h>
#include <stdint.h>

namespace {
constexpr int Nn = 128, Cc = 3, Hh = 224, Ww = 224;
constexpr int PAD = 4, HPp = 232, WPp = 232;
}

typedef __attribute__((ext_vector_type(4))) float v4f;

// One block per (image n, output row i). 256 threads (8 wave32).
// Phase 1: async-copy the two needed source rows (3 channels) into LDS
//          via global_load_async_to_lds_b128 (ASYNCcnt path).
// Phase 2: bilinear combine with reference-exact fp32 coordinate math,
//          non-temporal 16B stores (output is write-once).
__global__ __launch_bounds__(256) void rsa_kernel(const float* __restrict__ x,
                                                  const int* __restrict__ shift_h,
                                                  const int* __restrict__ shift_w,
                                                  float* __restrict__ out)
{
    __shared__ __align__(16) float lds[2][Cc][Ww];   // 2 rows x 3 ch x 224 f32 = 5376 B

    const int i = blockIdx.x;     // output row
    const int n = blockIdx.y;     // image
    const int t = threadIdx.x;

    const int sh = shift_h[n];
    const int sw = shift_w[n];

    // Replicate jnp.linspace(-1+1/hp, 1-1/hp, hp)[:h] in fp32.
    const float startc = (float)(-1.0 + 1.0 / 232.0);
    const float stopc  = (float)( 1.0 - 1.0 / 232.0);
    const float stepc  = (stopc - startc) / 231.0f;
    const float scl    = (float)(2.0 / 232.0);       // shift -> grid units

    // Row coordinate (uniform across the block -> scalar FP pipe).
    float gy  = startc + stepc * (float)i + (float)sh * scl;
    float iy  = ((gy + 1.0f) * 232.0f - 1.0f) * 0.5f;
    float y0f = floorf(iy);
    int   y0  = (int)y0f;
    float wy  = iy - y0f;
    int   y1  = y0 + 1;
    // Edge padding: xp[y][*] == x[clamp(y-PAD,0,H-1)][clamp(*-PAD,0,W-1)]
    int   ys0 = min(max(y0 - PAD, 0), Hh - 1);
    int   ys1 = min(max(y1 - PAD, 0), Hh - 1);

    // ---- Phase 1: async global -> LDS row staging ----
    const uint32_t lds_base = (uint32_t)(uintptr_t)(&lds[0][0][0]);
    const int nchunks = 2 * Cc * (Ww / 4);           // 336 x 16-byte chunks
    for (int idx = t; idx < nchunks; idx += 256) {
        int r   = idx / (Cc * (Ww / 4));
        int rem = idx - r * (Cc * (Ww / 4));
        int c   = rem / (Ww / 4);
        int q   = rem - c * (Ww / 4);                // 16B chunk within row
        int ys  = r ? ys1 : ys0;
        const float* gsrc = x + ((((size_t)n * Cc + c) * Hh + ys) * Ww + q * 4);
        uint64_t gaddr = (uint64_t)(uintptr_t)gsrc;
        uint32_t laddr = lds_base + (uint32_t)(((r * Cc + c) * Ww + q * 4) * sizeof(float));
        asm volatile("global_load_async_to_lds_b128 %0, %1, off"
                     :: "v"(laddr), "v"(gaddr)
                     : "memory");
    }
    asm volatile("s_wait_asynccnt 0" ::: "memory");
    __syncthreads();

    // ---- Phase 2: bilinear combine, 4 pixels/thread, NT 16B stores ----
    const float vy0 = (y0 >= 0 && y0 < HPp) ? 1.0f : 0.0f;
    const float vy1 = (y1 >= 0 && y1 < HPp) ? 1.0f : 0.0f;
    const float swf = (float)sw * scl;

    if (t < Cc * (Ww / 4)) {                         // 168 active lanes
        const int c  = t / (Ww / 4);
        const int j0 = (t - c * (Ww / 4)) * 4;
        v4f o;
#pragma unroll
        for (int u = 0; u < 4; ++u) {
            const int j = j0 + u;
            float gx  = startc + stepc * (float)j + swf;
            float ix  = ((gx + 1.0f) * 232.0f - 1.0f) * 0.5f;
            float x0f = floorf(ix);
            int   x0  = (int)x0f;
            float wx  = ix - x0f;
            int   x1  = x0 + 1;
            float vx0 = (x0 >= 0 && x0 < WPp) ? 1.0f : 0.0f;
            float vx1 = (x1 >= 0 && x1 < WPp) ? 1.0f : 0.0f;
            int xs0 = min(max(x0 - PAD, 0), Ww - 1);
            int xs1 = min(max(x1 - PAD, 0), Ww - 1);
            // validity folded into weights (exact: masks are 0.0f/1.0f)
            float v00 = lds[0][c][xs0] * (vx0 * vy0);
            float v01 = lds[0][c][xs1] * (vx1 * vy0);
            float v10 = lds[1][c][xs0] * (vx0 * vy1);
            float v11 = lds[1][c][xs1] * (vx1 * vy1);
            o[u] = v00 * (1.0f - wy) * (1.0f - wx)
                 + v01 * (1.0f - wy) * wx
                 + v10 * wy * (1.0f - wx)
                 + v11 * wy * wx;
        }
        v4f* dst = reinterpret_cast<v4f*>(
            out + ((((size_t)n * Cc + c) * Hh + i) * Ww + j0));
        __builtin_nontemporal_store(o, dst);         // keep L2 for the input rows
    }
}

extern "C" void kernel_launch(void* const* d_in, const int* in_sizes, int n_in,
                              void* d_out, int out_size, void* d_ws, size_t ws_size,
                              hipStream_t stream) {
    (void)in_sizes; (void)n_in; (void)d_ws; (void)ws_size; (void)out_size;
    const float* x  = (const float*)d_in[0];
    const int*   sh = (const int*)d_in[1];
    const int*   sw = (const int*)d_in[2];
    float*       o  = (float*)d_out;
    dim3 grid(Hh, Nn);   // 224 x 128 blocks
    rsa_kernel<<<grid, 256, 0, stream>>>(x, sh, sw, o);
}